// TopKNet_43138651521485
// MI455X (gfx1250) — compile-verified
//
#include <hip/hip_runtime.h>
#include <hip/hip_bf16.h>
#include <math.h>

// ---------------- problem constants (match reference) ----------------
#define BB   128          // graphs
#define NN   512          // nodes per graph
#define BN   65536        // BB*NN
#define CIN  64
#define HH   128
#define OUTC 10
#define DEG  8
#define EE   524288       // BN*DEG
#define KK1  410          // ceil(0.8*512)
#define KK2  328          // ceil(0.8*410)
#define KK3  263          // ceil(0.8*328)

typedef __attribute__((ext_vector_type(2))) float v2f;
typedef __attribute__((ext_vector_type(8))) float v8f;
typedef __attribute__((ext_vector_type(4))) int   v4i;

#define NEG_BIG (-3.402823466e38f)

// -------- gfx1250 async global->LDS staging (ASYNCcnt path) --------
#if __has_builtin(__builtin_amdgcn_global_load_async_to_lds_b128)
#define HAVE_ASYNC_LDS 1
#else
#define HAVE_ASYNC_LDS 0
#endif

__device__ __forceinline__ void async_cp16(const void* gp, void* lp) {
#if HAVE_ASYNC_LDS
    // Param 0 is a global-AS int4*; param 1 the LDS-AS int4*. AS1 shares the
    // generic VA; the AS3 offset is the low 32 bits of the flat LDS VA
    // (ISA section 10.2 aperture mapping), so integer laundering is value-correct.
    typedef __attribute__((address_space(1))) v4i* gptr_t;
    typedef __attribute__((address_space(3))) v4i* lptr_t;
    __builtin_amdgcn_global_load_async_to_lds_b128(
        (gptr_t)(unsigned long long)gp,
        (lptr_t)(unsigned int)(unsigned long long)lp,
        0, 0);
#else
    *(float4*)lp = *(const float4*)gp;
#endif
}

__device__ __forceinline__ void async_stage_wait() {
#if HAVE_ASYNC_LDS
#if __has_builtin(__builtin_amdgcn_s_wait_asynccnt)
    __builtin_amdgcn_s_wait_asynccnt(0);
#else
    asm volatile("s_wait_asynccnt 0" ::: "memory");
#endif
#endif
}

// ---------------- utility: zero fill ----------------
__global__ void zero_f32(float* __restrict__ p, long n) {
    long i = (long)blockIdx.x * blockDim.x + threadIdx.x;
    long stride = (long)gridDim.x * blockDim.x;
    for (; i < n; i += stride) p[i] = 0.0f;
}

// ---------------- scatter-add aggregation: agg[dst] += x[src] ----------------
// One thread per (edge, 4-feature chunk). Gated-dropped sources are exactly 0,
// so we can skip the atomics for them entirely (mask-free segment_sum).
__global__ void scatter_kernel(const float* __restrict__ xin,
                               const int* __restrict__ src,
                               const int* __restrict__ dst,
                               float* __restrict__ agg, int cq /* Cin/4 */) {
    long tid = (long)blockIdx.x * blockDim.x + threadIdx.x;
    long total = (long)EE * cq;
    if (tid >= total) return;
    int e = (int)(tid / cq);
    int c = (int)(tid % cq) * 4;
    int cin = cq * 4;
    int s = src[e], d = dst[e];
    const float* xp = xin + (size_t)s * cin + c;
    float v0 = xp[0], v1 = xp[1], v2 = xp[2], v3 = xp[3];
    if (v0 == 0.0f && v1 == 0.0f && v2 == 0.0f && v3 == 0.0f) return;
    float* ap = agg + (size_t)d * cin + c;
    __hip_atomic_fetch_add(ap + 0, v0, __ATOMIC_RELAXED, __HIP_MEMORY_SCOPE_AGENT);
    __hip_atomic_fetch_add(ap + 1, v1, __ATOMIC_RELAXED, __HIP_MEMORY_SCOPE_AGENT);
    __hip_atomic_fetch_add(ap + 2, v2, __ATOMIC_RELAXED, __HIP_MEMORY_SCOPE_AGENT);
    __hip_atomic_fetch_add(ap + 3, v3, __ATOMIC_RELAXED, __HIP_MEMORY_SCOPE_AGENT);
}

// ---------------- fused GraphConv GEMM via FP32 WMMA ----------------
// out = relu(agg @ Wrel + xin @ Wroot + bias), shapes [BN,Cin]@[Cin,128].
// One block = one 16-row M-tile; 8 waves cover the 8 N-tiles. Both 16xCin
// A-tiles (agg rows, x rows) are staged once into LDS with async b128 copies
// (shared by all 8 waves -> 8x less global A traffic), then each wave runs the
// 16x16x4 f32 WMMA K-loop from LDS.
// A (16x4 f32): lane l -> m=l&15, holds K = k0+(l>>4)*2 .. +1  (ISA 7.12.2)
// B (4x16 f32): lane l -> n=l&15, holds K = k0+(l>>4)*2 .. +1
// C/D (16x16 f32): VGPR r, lane l -> m = r+(l>>4)*8, n = l&15
__global__ void __launch_bounds__(256)
conv_wmma(const float* __restrict__ agg,
          const float* __restrict__ xin,
          const float* __restrict__ Wrel,
          const float* __restrict__ Wroot,
          const float* __restrict__ bias,
          float* __restrict__ out, int cin) {
    // row stride cin+12 floats: 16B-aligned rows (560B / 304B) and
    // (stride % 64) == 12 words -> 16 row-readers hit 16 distinct LDS banks.
    __shared__ __align__(16) float sA[16 * (HH + 12)];
    __shared__ __align__(16) float sX[16 * (HH + 12)];
    const int stride = cin + 12;

    int t = threadIdx.x;
    int R = blockIdx.x * 16;
    const float* gA = agg + (size_t)R * cin;   // 16 contiguous rows
    const float* gX = xin + (size_t)R * cin;

    int cq = cin >> 2;                          // float4 chunks per row
    for (int i = t; i < 16 * cq; i += 256) {
        int m = i / cq, c = (i % cq) * 4;
        async_cp16(gA + (size_t)m * cin + c, &sA[m * stride + c]);
        async_cp16(gX + (size_t)m * cin + c, &sX[m * stride + c]);
    }
    async_stage_wait();
    __syncthreads();

    int wave = t >> 5;
    int lane = t & 31;
    int Ncol = wave * 16;          // 8 waves == 8 column tiles of H=128
    int ml = lane & 15;
    int ksel = (lane >> 4) * 2;

    v8f acc = {};
    const float* ar = &sA[ml * stride];
    for (int k0 = 0; k0 < cin; k0 += 4) {
        v2f a, b;
        a.x = ar[k0 + ksel];
        a.y = ar[k0 + ksel + 1];
        b.x = Wrel[(size_t)(k0 + ksel) * HH + Ncol + ml];
        b.y = Wrel[(size_t)(k0 + ksel + 1) * HH + Ncol + ml];
        acc = __builtin_amdgcn_wmma_f32_16x16x4_f32(false, a, false, b,
                                                    (short)0, acc, false, false);
    }
    const float* xr = &sX[ml * stride];
    for (int k0 = 0; k0 < cin; k0 += 4) {
        v2f a, b;
        a.x = xr[k0 + ksel];
        a.y = xr[k0 + ksel + 1];
        b.x = Wroot[(size_t)(k0 + ksel) * HH + Ncol + ml];
        b.y = Wroot[(size_t)(k0 + ksel + 1) * HH + Ncol + ml];
        acc = __builtin_amdgcn_wmma_f32_16x16x4_f32(false, a, false, b,
                                                    (short)0, acc, false, false);
    }
    float bv = bias[Ncol + ml];    // this lane's n-column bias
#pragma unroll
    for (int r = 0; r < 8; r++) {
        int m = r + (lane >> 4) * 8;
        float v = acc[r] + bv;
        v = v > 0.0f ? v : 0.0f;
        out[(size_t)(R + m) * HH + Ncol + ml] = v;
    }
}

// ---------------- ||p|| ----------------
__global__ void pnorm_kernel(const float* __restrict__ p, float* __restrict__ out) {
    __shared__ float red[HH];
    int t = threadIdx.x;
    float v = p[t];
    red[t] = v * v;
    __syncthreads();
    for (int s = HH / 2; s > 0; s >>= 1) {
        if (t < s) red[t] += red[t + s];
        __syncthreads();
    }
    if (t == 0) out[0] = sqrtf(red[0]);
}

// ---------------- score = tanh(h . p / ||p||), one wave per node ----------------
__global__ void score_kernel(const float* __restrict__ h,
                             const float* __restrict__ p,
                             const float* __restrict__ pnorm,
                             float* __restrict__ score) {
    int wid = threadIdx.x >> 5;
    int lane = threadIdx.x & 31;
    int node = blockIdx.x * (blockDim.x >> 5) + wid;
    if (node >= BN) return;
    const float* row = h + (size_t)node * HH;
    float s = 0.0f;
#pragma unroll
    for (int c = 0; c < HH / 32; c++) s += row[lane + c * 32] * p[lane + c * 32];
#pragma unroll
    for (int off = 16; off > 0; off >>= 1) s += __shfl_xor(s, off, 32);
    if (lane == 0) score[node] = tanhf(s / pnorm[0]);
}

// ---------------- exact top-k per graph (matches jax.lax.top_k tie rules) ----
// One block (512 threads) per graph; bitonic sort (score desc, idx asc) in LDS.
// sel doubles as nmask of the previous stage: read first, write after barrier.
__global__ void topk_kernel(const float* __restrict__ score,
                            int* __restrict__ sel, int K, int use_mask) {
    __shared__ float ss[NN];
    __shared__ int ii[NN];
    int t = threadIdx.x;
    int base = blockIdx.x * NN;
    float sc = score[base + t];
    if (use_mask && sel[base + t] == 0) sc = NEG_BIG;
    ss[t] = sc;
    ii[t] = t;
    __syncthreads();
    for (int k = 2; k <= NN; k <<= 1) {
        for (int j = k >> 1; j > 0; j >>= 1) {
            int ixj = t ^ j;
            if (ixj > t) {
                float a = ss[t], b = ss[ixj];
                int ia = ii[t], ib = ii[ixj];
                bool a_first = (a > b) || (a == b && ia < ib);
                bool desc = ((t & k) == 0);
                if (desc != a_first) {
                    ss[t] = b; ss[ixj] = a;
                    ii[t] = ib; ii[ixj] = ia;
                }
            }
            __syncthreads();
        }
    }
    sel[base + ii[t]] = (t < K) ? 1 : 0;
}

// ---------------- gate: h *= sel ? score : 0 ----------------
__global__ void gate_kernel(float* __restrict__ h,
                            const float* __restrict__ score,
                            const int* __restrict__ sel) {
    long i = (long)blockIdx.x * blockDim.x + threadIdx.x;
    if (i >= (long)BN * HH) return;
    int node = (int)(i >> 7);   // /HH
    float g = sel[node] ? score[node] : 0.0f;
    h[i] *= g;
}

// ---------------- readout: z += [max | mean] over selected nodes -------------
__global__ void readout_kernel(const float* __restrict__ h,
                               const int* __restrict__ sel,
                               float* __restrict__ z, float invK) {
    int b = blockIdx.x;
    int f = threadIdx.x;   // HH threads
    float mx = NEG_BIG, sm = 0.0f;
    for (int n = 0; n < NN; n++) {
        int node = b * NN + n;
        if (sel[node]) {
            float v = h[(size_t)node * HH + f];
            mx = fmaxf(mx, v);
            sm += v;
        }
    }
    z[(size_t)b * 2 * HH + f]      += mx;
    z[(size_t)b * 2 * HH + HH + f] += sm * invK;
}

// ---------------- small MLP layers ----------------
__global__ void linear_kernel(const float* __restrict__ in,
                              const float* __restrict__ w,
                              const float* __restrict__ bias,
                              float* __restrict__ out,
                              int I, int O, int do_relu) {
    int row = blockIdx.x;
    int j = threadIdx.x;
    if (j >= O) return;
    float s = bias[j];
    const float* ir = in + (size_t)row * I;
    for (int i = 0; i < I; i++) s += ir[i] * w[(size_t)i * O + j];
    if (do_relu) s = s > 0.0f ? s : 0.0f;
    out[(size_t)row * O + j] = s;
}

__global__ void head_kernel(const float* __restrict__ in /*[B,64]*/,
                            const float* __restrict__ w /*[64,10]*/,
                            const float* __restrict__ bias,
                            float* __restrict__ out /*[B,10]*/) {
    int row = blockIdx.x * blockDim.x + threadIdx.x;
    if (row >= BB) return;
    const float* ir = in + (size_t)row * 64;
    float logits[OUTC];
#pragma unroll
    for (int j = 0; j < OUTC; j++) {
        float s = bias[j];
        for (int i = 0; i < 64; i++) s += ir[i] * w[i * OUTC + j];
        logits[j] = s;
    }
    float m = logits[0];
#pragma unroll
    for (int j = 1; j < OUTC; j++) m = fmaxf(m, logits[j]);
    float sum = 0.0f;
#pragma unroll
    for (int j = 0; j < OUTC; j++) sum += expf(logits[j] - m);
    float lse = m + logf(sum);
#pragma unroll
    for (int j = 0; j < OUTC; j++) out[(size_t)row * OUTC + j] = logits[j] - lse;
}

// ---------------- host orchestration ----------------
extern "C" void kernel_launch(void* const* d_in, const int* in_sizes, int n_in,
                              void* d_out, int out_size, void* d_ws, size_t ws_size,
                              hipStream_t stream) {
    (void)in_sizes; (void)n_in; (void)out_size; (void)ws_size;

    const float* x       = (const float*)d_in[0];
    const int*   ei      = (const int*)d_in[1];
    const int*   src     = ei;
    const int*   dst     = ei + EE;
    const float* W1_rel  = (const float*)d_in[3];
    const float* W1_root = (const float*)d_in[4];
    const float* b1      = (const float*)d_in[5];
    const float* p1      = (const float*)d_in[6];
    const float* W2_rel  = (const float*)d_in[7];
    const float* W2_root = (const float*)d_in[8];
    const float* b2      = (const float*)d_in[9];
    const float* p2      = (const float*)d_in[10];
    const float* W3_rel  = (const float*)d_in[11];
    const float* W3_root = (const float*)d_in[12];
    const float* b3      = (const float*)d_in[13];
    const float* p3      = (const float*)d_in[14];
    const float* lin1_w  = (const float*)d_in[15];
    const float* lin1_b  = (const float*)d_in[16];
    const float* lin2_w  = (const float*)d_in[17];
    const float* lin2_b  = (const float*)d_in[18];
    const float* lin3_w  = (const float*)d_in[19];
    const float* lin3_b  = (const float*)d_in[20];

    // workspace layout (floats)
    float* ws = (float*)d_ws;
    const size_t NH = (size_t)BN * HH;      // 8.39M floats
    float* hA    = ws;                       // [BN,H]
    float* hB    = hA + NH;                  // [BN,H]
    float* agg   = hB + NH;                  // [BN,H] (stage1 uses BN*64)
    float* score = agg + NH;                 // [BN]
    int*   sel   = (int*)(score + BN);       // [BN] (sel of current == nmask of next)
    float* z     = (float*)(sel + BN);       // [B, 2H]
    float* pn    = z + (size_t)BB * 2 * HH;  // [1]
    float* t1    = pn + 16;                  // [B, H]
    float* t2    = t1 + (size_t)BB * HH;     // [B, 64]

    // zero the readout accumulator
    zero_f32<<<256, 256, 0, stream>>>(z, (long)BB * 2 * HH);

    struct Stage {
        const float* xin; float* hout; int cin;
        const float* Wrel; const float* Wroot; const float* bias; const float* p;
        int K; int use_mask;
    };
    Stage stages[3] = {
        { x,  hA, CIN, W1_rel, W1_root, b1, p1, KK1, 0 },
        { hA, hB, HH,  W2_rel, W2_root, b2, p2, KK2, 1 },
        { hB, hA, HH,  W3_rel, W3_root, b3, p3, KK3, 1 },
    };

    for (int s = 0; s < 3; s++) {
        Stage& st = stages[s];
        int cq = st.cin / 4;

        // agg = 0
        zero_f32<<<2048, 256, 0, stream>>>(agg, (long)BN * st.cin);
        // agg[dst] += xin[src]  (mask-free: gated-dropped sources are exactly 0)
        long sthreads = (long)EE * cq;
        int sblocks = (int)((sthreads + 255) / 256);
        scatter_kernel<<<sblocks, 256, 0, stream>>>(st.xin, src, dst, agg, cq);
        // h = relu(agg@Wrel + xin@Wroot + b)  -- FP32 WMMA, LDS-staged A tiles
        conv_wmma<<<BN / 16, 256, 0, stream>>>(agg, st.xin, st.Wrel,
                                               st.Wroot, st.bias,
                                               st.hout, st.cin);
        // ||p||, scores, exact top-k, gating
        pnorm_kernel<<<1, HH, 0, stream>>>(st.p, pn);
        score_kernel<<<BN / 8, 256, 0, stream>>>(st.hout, st.p, pn, score);
        topk_kernel<<<BB, NN, 0, stream>>>(score, sel, st.K, st.use_mask);
        gate_kernel<<<(int)(NH / 256), 256, 0, stream>>>(st.hout, score, sel);
        // z += [gmax | gmean]
        readout_kernel<<<BB, HH, 0, stream>>>(st.hout, sel, z, 1.0f / (float)st.K);
    }

    // MLP head
    linear_kernel<<<BB, HH, 0, stream>>>(z, lin1_w, lin1_b, t1, 2 * HH, HH, 1);
    linear_kernel<<<BB, 64, 0, stream>>>(t1, lin2_w, lin2_b, t2, HH, 64, 1);
    head_kernel<<<1, BB, 0, stream>>>(t2, lin3_w, lin3_b, (float*)d_out);
}